// YoloLoss_76527727280784
// MI455X (gfx1250) — compile-verified
//
#include <hip/hip_runtime.h>

typedef float v2f __attribute__((ext_vector_type(2)));
typedef float v8f __attribute__((ext_vector_type(8)));

#define LCOORD 8.0f
#define LNOOBJ 1.0f
#define LCLASS 0.7f
#define EPSF   1e-10f

__device__ __forceinline__ float sq(float x) { return x * x; }

// boxes are (x, y, sqrt(w), sqrt(h), conf); center-format IoU, EPS in denom
__device__ __forceinline__ float iou5(const float* b1, const float* b2) {
  float w1 = b1[2] * b1[2], h1 = b1[3] * b1[3];
  float w2 = b2[2] * b2[2], h2 = b2[3] * b2[3];
  float left  = fmaxf(b1[0] - 0.5f * w1, b2[0] - 0.5f * w2);
  float right = fminf(b1[0] + 0.5f * w1, b2[0] + 0.5f * w2);
  float top   = fmaxf(b1[1] - 0.5f * h1, b2[1] - 0.5f * h2);
  float bot   = fminf(b1[1] + 0.5f * h1, b2[1] + 0.5f * h2);
  float inter = fmaxf(right - left, 0.f) * fmaxf(bot - top, 0.f);
  float uni   = w1 * h1 + w2 * h2 - inter;
  return inter / (uni + EPSF);
}

// Stage 1: streaming pass, one cell (14 floats x 2 tensors = 112 B) per thread
// per grid-stride step. Deterministic block-local LDS tree -> partial[block].
__global__ void yolo_partial(const float* __restrict__ outp,
                             const float* __restrict__ gtp,
                             float* __restrict__ partial, int ncells) {
  const int tid    = threadIdx.x;
  const int stride = gridDim.x * blockDim.x;
  float acc = 0.f;

  for (int cell = blockIdx.x * blockDim.x + tid; cell < ncells; cell += stride) {
    float o[14], g[14];
    const float2* o2 = reinterpret_cast<const float2*>(outp) + (size_t)cell * 7;
    const float2* g2 = reinterpret_cast<const float2*>(gtp)  + (size_t)cell * 7;
#pragma unroll
    for (int i = 0; i < 7; ++i) { float2 t = o2[i]; o[2*i] = t.x; o[2*i+1] = t.y; }
#pragma unroll
    for (int i = 0; i < 7; ++i) { float2 t = g2[i]; g[2*i] = t.x; g[2*i+1] = t.y; }

    float obj   = (g[4] > 0.f) ? 1.f : 0.f;
    float noobj = 1.f - obj;

    float iou0 = iou5(o, g);       // box 0 vs gt box 0
    float iou1 = iou5(o + 5, g);   // box 1 vs gt box 0
    bool  second  = (iou1 > iou0); // jnp.argmax: first index wins ties
    float max_iou = fmaxf(iou0, iou1);

    // select responsible / other box with cndmask (keep arrays in VGPRs)
    float pr0 = second ? o[5] : o[0];
    float pr1 = second ? o[6] : o[1];
    float pr2 = second ? o[7] : o[2];
    float pr3 = second ? o[8] : o[3];
    float pr4 = second ? o[9] : o[4];
    float po4 = second ? o[4] : o[9];
    float gr0 = second ? g[5] : g[0];
    float gr1 = second ? g[6] : g[1];
    float gr2 = second ? g[7] : g[2];
    float gr3 = second ? g[8] : g[3];
    float go4 = second ? g[4] : g[9];

    float no_l = noobj * (sq(o[4] - g[4]) + sq(o[9] - g[9]));
    float conf = obj * sq(pr4 - max_iou);
    float loc  = obj * (sq(pr0 - gr0) + sq(pr1 - gr1) + sq(pr2 - gr2) + sq(pr3 - gr3));
    float nbc  = obj * sq(po4 - go4);
    float cls  = obj * (sq(o[10] - g[10]) + sq(o[11] - g[11]) +
                        sq(o[12] - g[12]) + sq(o[13] - g[13]));

    acc += LNOOBJ * no_l + conf + LCOORD * loc + LNOOBJ * nbc + LCLASS * cls;
  }

  __shared__ float sdata[256];
  sdata[tid] = acc;
  __syncthreads();
#pragma unroll
  for (int s = 128; s > 0; s >>= 1) {
    if (tid < s) sdata[tid] += sdata[tid + s];
    __syncthreads();
  }
  if (tid == 0) partial[blockIdx.x] = sdata[0];
}

// Stage 2: one wave32. Each lane accumulates two strided partials (q_lane,
// q_lane+32); the 64 values are packed into the B operand of a single
// v_wmma_f32_16x16x4_f32 with A = ones, so D[m][n] = sum_k B[k][n] collapses
// 64 -> 16 column sums in one matrix op. Four xor-shuffles finish 16 -> 1.
// (A = ones makes the total independent of the exact B lane layout.)
__global__ void yolo_final(const float* __restrict__ ws, float* __restrict__ res,
                           int nb, float inv_n) {
  const int lane = threadIdx.x; // 0..31, single wave, EXEC all-ones at WMMA
  float p0 = 0.f, p1 = 0.f;
  for (int j = lane;      j < nb; j += 64) p0 += ws[j];
  for (int j = lane + 32; j < nb; j += 64) p1 += ws[j];

  v2f a; a.x = 1.0f; a.y = 1.0f;   // A (16x4) all-ones
  v2f b; b.x = p0;   b.y = p1;     // B (4x16) holds all 64 partials
  v8f c = {};
  // 8 args: (neg_a, A, neg_b, B, c_mod, C, reuse_a, reuse_b)
  c = __builtin_amdgcn_wmma_f32_16x16x4_f32(false, a, false, b, (short)0, c,
                                            false, false);
  float v = c[0];                  // = column-sum C_{lane%16}
  v += __shfl_xor(v, 1, 32);
  v += __shfl_xor(v, 2, 32);
  v += __shfl_xor(v, 4, 32);
  v += __shfl_xor(v, 8, 32);
  if (lane == 0) res[0] = v * inv_n;
}

extern "C" void kernel_launch(void* const* d_in, const int* in_sizes, int n_in,
                              void* d_out, int out_size, void* d_ws, size_t ws_size,
                              hipStream_t stream) {
  const float* outp = (const float*)d_in[0];
  const float* gtp  = (const float*)d_in[1];
  float* ws  = (float*)d_ws;
  float* res = (float*)d_out;

  const int total  = in_sizes[0];      // N*H*W*C = N*672
  const int ncells = total / 14;       // N*H*W
  const int threads = 256;
  int blocks = 2048;                   // grid-stride; keeps ws tiny (8 KB)
  int maxb = (ncells + threads - 1) / threads;
  if (blocks > maxb) blocks = maxb;
  const float inv_n = 672.0f / (float)total;  // 1/N

  yolo_partial<<<blocks, threads, 0, stream>>>(outp, gtp, ws, ncells);
  yolo_final<<<1, 32, 0, stream>>>(ws, res, blocks, inv_n);
}